// ProposalLayer_82360292868834
// MI455X (gfx1250) — compile-verified
//
#include <hip/hip_runtime.h>
#include <math.h>

// Problem constants (match reference)
#define BATCH      8
#define NANCH      261888      // = 1023 * 256
#define PRE_NMS    6000
#define NUM_PROP   2000
#define SEL_PAD    6144        // PRE_NMS padded for alignment
#define SORT_N     8192        // bitonic size
#define NBINS      1024
#define NMS_THR    0.7f

// ---------------- score binning helpers (must be bit-identical across kernels) --------
__device__ __forceinline__ int bin_of(float s) {
    int b = (int)(s * 1024.0f);
    return b < 0 ? 0 : (b > 1023 ? 1023 : b);
}
__device__ __forceinline__ int sub_of(float s, int bin) {
    float f = s * 1024.0f - (float)bin;
    int u = (int)(f * 1024.0f);
    return u < 0 ? 0 : (u > 1023 ? 1023 : u);
}

// ---------------- CDNA5 async global->LDS copy (CDNA5 ISA) -----------
// Per-lane 16B copy global->LDS, tracked on ASYNCcnt. VDST VGPR holds the LDS byte
// address (low 32 bits of the generic shared pointer), VADDR the 64-bit global address.
__device__ __forceinline__ void async_load_lds_b128(unsigned lds_addr,
                                                    unsigned long long gaddr) {
    asm volatile("global_load_async_to_lds_b128 %0, %1, off"
                 :: "v"(lds_addr), "v"(gaddr)
                 : "memory");
}
__device__ __forceinline__ void wait_asynccnt0() {
    asm volatile("s_wait_asynccnt 0" ::: "memory");
}

// ---------------- kernel 1: per-batch 1024-bin histogram of scores -------------------
__global__ void pl_hist1(const float* __restrict__ cp, unsigned* __restrict__ hist) {
    __shared__ unsigned lh[NBINS];
    const int tid = threadIdx.x;
    for (int k = tid; k < NBINS; k += 256) lh[k] = 0u;
    __syncthreads();
    const int b = blockIdx.y;
    const int i = blockIdx.x * 256 + tid;            // grid.x*256 == NANCH exactly
    const float s = cp[((size_t)b * NANCH + i) * 2 + 1];
    atomicAdd(&lh[bin_of(s)], 1u);
    __syncthreads();
    for (int k = tid; k < NBINS; k += 256)
        if (lh[k]) atomicAdd(&hist[b * NBINS + k], lh[k]);
}

// ---------------- kernel 2: find coarse threshold bin --------------------------------
__global__ void pl_thresh1(const unsigned* __restrict__ hist, unsigned* __restrict__ binfo) {
    if (threadIdx.x != 0) return;
    const int b = blockIdx.x;
    const unsigned* h = hist + b * NBINS;
    unsigned cum = 0; int t = NBINS;
    do { --t; cum += h[t]; } while (cum < PRE_NMS && t > 0);
    binfo[b * 8 + 0] = (unsigned)t;                   // t1
    binfo[b * 8 + 1] = cum - h[t];                    // countAbove (strictly above bin t1)
}

// ---------------- kernel 3: sub-histogram inside the threshold bin -------------------
__global__ void pl_hist2(const float* __restrict__ cp, const unsigned* __restrict__ binfo,
                         unsigned* __restrict__ hist2) {
    __shared__ unsigned lh[NBINS];
    const int tid = threadIdx.x;
    for (int k = tid; k < NBINS; k += 256) lh[k] = 0u;
    __syncthreads();
    const int b = blockIdx.y;
    const unsigned t1 = binfo[b * 8 + 0];
    const int i = blockIdx.x * 256 + tid;
    const float s = cp[((size_t)b * NANCH + i) * 2 + 1];
    const int bin = bin_of(s);
    if ((unsigned)bin == t1) atomicAdd(&lh[sub_of(s, bin)], 1u);
    __syncthreads();
    for (int k = tid; k < NBINS; k += 256)
        if (lh[k]) atomicAdd(&hist2[b * NBINS + k], lh[k]);
}

// ---------------- kernel 4: fine threshold + boundary quota --------------------------
__global__ void pl_thresh2(const unsigned* __restrict__ hist2, unsigned* __restrict__ binfo) {
    if (threadIdx.x != 0) return;
    const int b = blockIdx.x;
    const unsigned* h = hist2 + b * NBINS;
    const unsigned need = PRE_NMS - binfo[b * 8 + 1];
    unsigned cum = 0; int t = NBINS;
    do { --t; cum += h[t]; } while (cum < need && t > 0);
    binfo[b * 8 + 2] = (unsigned)t;                   // t2
    binfo[b * 8 + 3] = need - (cum - h[t]);           // needInBin (>=1 by construction)
}

// ---------------- pad kernel: -inf keys for sort padding -----------------------------
__global__ void pl_pad(float* __restrict__ sel_scores) {
    const int b = blockIdx.x;
    sel_scores[(size_t)b * SEL_PAD + PRE_NMS + threadIdx.x] = -1e30f;
}

// ---------------- kernel 5: select top-6000, decode + clip boxes ---------------------
__global__ void pl_compact(const float* __restrict__ cp, const float* __restrict__ dlt,
                           const float* __restrict__ anc,
                           const unsigned* __restrict__ binfo, unsigned* __restrict__ counters,
                           float* __restrict__ sel_scores, float* __restrict__ sel_boxes) {
    const int b = blockIdx.y;
    const int i = blockIdx.x * 256 + threadIdx.x;
    const size_t base = (size_t)b * NANCH + i;
    const float s = cp[base * 2 + 1];
    const int bin = bin_of(s);
    const unsigned t1 = binfo[b * 8 + 0];
    if ((unsigned)bin < t1) return;
    if ((unsigned)bin == t1) {
        const int sub = sub_of(s, bin);
        const unsigned t2 = binfo[b * 8 + 2];
        if ((unsigned)sub < t2) return;
        if ((unsigned)sub == t2) {
            const unsigned quota  = binfo[b * 8 + 3];
            const unsigned ticket = atomicAdd(&counters[b * 4 + 1], 1u);
            if (ticket >= quota) return;              // exact top-k at the boundary value
        }
    }
    const unsigned pos = atomicAdd(&counters[b * 4 + 0], 1u);   // < PRE_NMS guaranteed

    // decode: deltas * BBOX_STD, center shift, exp h/w, clip [0,1]
    const float d0 = dlt[base * 4 + 0] * 0.1f;
    const float d1 = dlt[base * 4 + 1] * 0.1f;
    const float d2 = dlt[base * 4 + 2] * 0.2f;
    const float d3 = dlt[base * 4 + 3] * 0.2f;
    const float a0 = anc[base * 4 + 0], a1 = anc[base * 4 + 1];
    const float a2 = anc[base * 4 + 2], a3 = anc[base * 4 + 3];
    float h = a2 - a0, w = a3 - a1;
    const float cy = a0 + 0.5f * h + d0 * h;
    const float cx = a1 + 0.5f * w + d1 * w;
    h *= expf(d2);
    w *= expf(d3);
    float4 bx;
    bx.x = fminf(fmaxf(cy - 0.5f * h, 0.f), 1.f);
    bx.y = fminf(fmaxf(cx - 0.5f * w, 0.f), 1.f);
    bx.z = fminf(fmaxf(cy + 0.5f * h, 0.f), 1.f);
    bx.w = fminf(fmaxf(cx + 0.5f * w, 0.f), 1.f);
    *(float4*)(sel_boxes + ((size_t)b * SEL_PAD + pos) * 4) = bx;
    sel_scores[(size_t)b * SEL_PAD + pos] = s;
}

// ---------------- kernel 6: per-batch bitonic sort (desc) + LDS permute --------------
// Boxes are async-DMA'd into LDS while the bitonic sort runs (latency hiding on
// ASYNCcnt), then the permute-gather reads LDS instead of random global memory.
// LDS: keys 32KB + vals 32KB + boxes 96KB = 160KB (fits CDNA5 320KB WGP LDS).
__global__ void __launch_bounds__(1024)
pl_sort(const float* __restrict__ sel_scores, const float* __restrict__ sel_boxes,
        float* __restrict__ sorted_boxes) {
    extern __shared__ float smem[];
    float* keys   = smem;                       // [8192]
    int*   vals   = (int*)(smem + SORT_N);      // [8192]
    float* boxlds = smem + 2 * SORT_N;          // [SEL_PAD*4]
    const int b = blockIdx.x, tid = threadIdx.x;

    // kick off async staging of the (unsorted) box table into LDS
    const unsigned box_base = (unsigned)(size_t)(void*)boxlds;  // low 32 bits = LDS addr
    const float* bsrc = sel_boxes + (size_t)b * SEL_PAD * 4;
    for (int c = tid; c < SEL_PAD; c += 1024)
        async_load_lds_b128(box_base + (unsigned)c * 16u,
                            (unsigned long long)(size_t)(bsrc + (size_t)c * 4));

    for (int k = tid; k < SORT_N; k += 1024) {
        keys[k] = (k < SEL_PAD) ? sel_scores[(size_t)b * SEL_PAD + k] : -1e30f;
        vals[k] = k;
    }
    for (int size = 2; size <= SORT_N; size <<= 1) {
        for (int stride = size >> 1; stride > 0; stride >>= 1) {
            __syncthreads();
            for (int i = tid; i < SORT_N; i += 1024) {
                const int j = i ^ stride;
                if (j > i) {
                    const bool desc = ((i & size) == 0);
                    const float ki = keys[i], kj = keys[j];
                    if (desc ? (ki < kj) : (ki > kj)) {
                        keys[i] = kj; keys[j] = ki;
                        const int tv = vals[i]; vals[i] = vals[j]; vals[j] = tv;
                    }
                }
            }
        }
    }
    wait_asynccnt0();                 // per-wave async completion, then block-wide sync
    __syncthreads();
    for (int k = tid; k < PRE_NMS; k += 1024) {
        const int idx = vals[k];
        const float4 bx = ((const float4*)boxlds)[idx];
        *(float4*)(sorted_boxes + ((size_t)b * PRE_NMS + k) * 4) = bx;
    }
}

// ---------------- kernel 7: greedy NMS, all state in LDS -----------------------------
// Sorted-order greedy NMS == reference iterative-argmax NMS. Thread 0 scans the alive
// bitmask to the next live box, so the loop runs once per *selection* (<= 2001 iters,
// 2 barriers each) instead of once per candidate. All j <= i are provably dead, so
// suppression only covers j > i.
__global__ void __launch_bounds__(1024)
pl_nms(const float* __restrict__ sorted_boxes, float* __restrict__ out) {
    extern __shared__ float smem[];
    float*    boxes = smem;                           // 24000 f32 (96 KB)
    unsigned* alive = (unsigned*)(smem + PRE_NMS * 4);// 192 words
    __shared__ int   sh_flag, sh_out, sh_i;
    __shared__ float sh_box[4];

    const int b = blockIdx.x, tid = threadIdx.x;
    const float* src = sorted_boxes + (size_t)b * PRE_NMS * 4;

    // Stage the box table into LDS via CDNA5 async global->LDS b128 copies.
    const unsigned lds_base = (unsigned)(size_t)(void*)boxes;   // low 32 bits = LDS addr
    for (int c = tid; c < PRE_NMS; c += 1024)
        async_load_lds_b128(lds_base + (unsigned)c * 16u,
                            (unsigned long long)(size_t)(src + (size_t)c * 4));

    for (int w = tid; w < 192; w += 1024)
        alive[w] = (w < 187) ? 0xFFFFFFFFu : ((w == 187) ? 0x0000FFFFu : 0u); // 6000 bits
    if (tid == 0) { sh_flag = 0; sh_out = 0; sh_i = 0; }
    wait_asynccnt0();
    __syncthreads();

    int pos = 0;                                      // thread 0's persistent scan cursor
    while (true) {
        if (tid == 0) {
            int found = -1;
            if (sh_out < NUM_PROP) {
                int w = pos >> 5;
                unsigned word = (w < 188) ? (alive[w] & (0xFFFFFFFFu << (pos & 31))) : 0u;
                while (true) {
                    if (word) { found = (w << 5) + __ffs(word) - 1; break; }
                    if (++w >= 188) break;
                    word = alive[w];
                }
            }
            if (found < 0) {
                sh_flag = -1;
            } else {
                const float4 bx = ((const float4*)boxes)[found];
                sh_box[0] = bx.x; sh_box[1] = bx.y; sh_box[2] = bx.z; sh_box[3] = bx.w;
                const int o = sh_out;
                float* dst = out + ((size_t)b * NUM_PROP + o) * 4;
                dst[0] = bx.x; dst[1] = bx.y; dst[2] = bx.z; dst[3] = bx.w;
                sh_out = o + 1;
                alive[found >> 5] &= ~(1u << (found & 31));  // remove selected
                sh_i = found;
                sh_flag = 1;
                pos = found + 1;
            }
        }
        __syncthreads();
        if (sh_flag < 0) break;
        const int i = sh_i;
        const float by1 = sh_box[0], bx1 = sh_box[1], by2 = sh_box[2], bx2 = sh_box[3];
        const float areaB = (by2 - by1) * (bx2 - bx1);
        for (int j = i + 1 + tid; j < PRE_NMS; j += 1024) {
            const unsigned wbits = alive[j >> 5];
            if (!((wbits >> (j & 31)) & 1)) continue;
            const float q0 = boxes[4 * j + 0], q1 = boxes[4 * j + 1];
            const float q2 = boxes[4 * j + 2], q3 = boxes[4 * j + 3];
            const float y1 = fmaxf(by1, q0), x1 = fmaxf(bx1, q1);
            const float y2 = fminf(by2, q2), x2 = fminf(bx2, q3);
            const float inter = fmaxf(y2 - y1, 0.f) * fmaxf(x2 - x1, 0.f);
            const float a2v = (q2 - q0) * (q3 - q1);
            const float iou = inter / (areaB + a2v - inter + 1e-9f);
            if (iou > NMS_THR) atomicAnd(&alive[j >> 5], ~(1u << (j & 31)));
        }
        __syncthreads();   // alive updates visible before thread 0's next scan/write
    }
    // zero-pad the tail (reference emits zeros once no valid boxes remain)
    const int cnt = sh_out;
    for (int k = cnt + tid; k < NUM_PROP; k += 1024) {
        float* dst = out + ((size_t)b * NUM_PROP + k) * 4;
        dst[0] = 0.f; dst[1] = 0.f; dst[2] = 0.f; dst[3] = 0.f;
    }
}

// ---------------- host launch --------------------------------------------------------
extern "C" void kernel_launch(void* const* d_in, const int* in_sizes, int n_in,
                              void* d_out, int out_size, void* d_ws, size_t ws_size,
                              hipStream_t stream) {
    const float* cp  = (const float*)d_in[0];   // class_probs (B,N,2)
    const float* dlt = (const float*)d_in[1];   // bbox_offset (B,N,4)
    const float* anc = (const float*)d_in[2];   // anchors     (B,N,4)
    float* out = (float*)d_out;

    char* ws = (char*)d_ws;
    unsigned* hist1    = (unsigned*)(ws + 0);          //  32768 B
    unsigned* hist2    = (unsigned*)(ws + 32768);      //  32768 B
    unsigned* counters = (unsigned*)(ws + 65536);      //    256 B (pos,ticket per batch)
    unsigned* binfo    = (unsigned*)(ws + 65792);      //    256 B
    float* sel_scores   = (float*)(ws + 66048);        // 196608 B
    float* sel_boxes    = (float*)(ws + 262656);       // 786432 B
    float* sorted_boxes = (float*)(ws + 1049088);      // 768000 B  (total ~1.74 MB)

    hipMemsetAsync(ws, 0, 66048, stream);              // hists + counters + binfo

    dim3 gridN(NANCH / 256, BATCH);
    pl_pad    <<<BATCH, SEL_PAD - PRE_NMS, 0, stream>>>(sel_scores);
    pl_hist1  <<<gridN, 256, 0, stream>>>(cp, hist1);
    pl_thresh1<<<BATCH, 1,   0, stream>>>(hist1, binfo);
    pl_hist2  <<<gridN, 256, 0, stream>>>(cp, binfo, hist2);
    pl_thresh2<<<BATCH, 1,   0, stream>>>(hist2, binfo);
    pl_compact<<<gridN, 256, 0, stream>>>(cp, dlt, anc, binfo, counters,
                                          sel_scores, sel_boxes);
    pl_sort<<<BATCH, 1024, (2 * SORT_N) * 4 + SEL_PAD * 16, stream>>>(
        sel_scores, sel_boxes, sorted_boxes);
    pl_nms <<<BATCH, 1024, PRE_NMS * 16 + 192 * 4, stream>>>(sorted_boxes, out);
}